// SimpleGraphAttention_15204184227952
// MI455X (gfx1250) — compile-verified
//
#include <hip/hip_runtime.h>
#include <math.h>

#define B_  8
#define N_  2048
#define D_  32
#define H_  4
#define HD_ 8
#define WAVES_PER_BLOCK 4

typedef float v2f __attribute__((ext_vector_type(2)));
typedef float v8f __attribute__((ext_vector_type(8)));

// ---------------------------------------------------------------------------
// Kernel 1: qkv = x @ Wqkv + bqkv, scattered into WMMA-friendly layouts:
//   q_ws, k_ws : [B*H, N, 8]   (row-major per head)
//   vT_ws      : [B*H, 8, N]   (head-dim major -> float2 B-fragment loads)
// One wave32 per row of x; x value broadcast via shuffles.
// ---------------------------------------------------------------------------
__global__ __launch_bounds__(256) void qkv_kernel(
    const float* __restrict__ x, const float* __restrict__ Wqkv,
    const float* __restrict__ bqkv,
    float* __restrict__ q_ws, float* __restrict__ k_ws, float* __restrict__ vT_ws)
{
    const int row = blockIdx.x * 8 + threadIdx.y;   // [0, B*N)
    const int t   = threadIdx.x;                    // 0..31
    const int b   = row / N_;
    const int n   = row % N_;

    const float xv = x[(size_t)row * D_ + t];
    float a0 = bqkv[t];
    float a1 = bqkv[t + 32];
    float a2 = bqkv[t + 64];
#pragma unroll
    for (int d = 0; d < 32; ++d) {
        const float xd = __shfl(xv, d, 32);
        a0 = fmaf(xd, Wqkv[d * 96 + t],      a0);
        a1 = fmaf(xd, Wqkv[d * 96 + 32 + t], a1);
        a2 = fmaf(xd, Wqkv[d * 96 + 64 + t], a2);
    }
    const int h = t >> 3, hd = t & 7;
    const int bh = b * H_ + h;
    q_ws [((size_t)bh * N_ + n) * HD_ + hd] = a0;
    k_ws [((size_t)bh * N_ + n) * HD_ + hd] = a1;
    vT_ws[((size_t)bh * HD_ + hd) * N_ + n] = a2;
}

// ---------------------------------------------------------------------------
// Kernel 2: fused attention with adjacency-prior blending.
// One wave32 per (b,h, 16-row query tile). Streams 128 key tiles of 16.
//   S = Q·K^T            via 2x V_WMMA_F32_16X16X4_F32 (K=8)
//   P = exp(a1*S+a2*adj) (shift-free softmax numerator; scores are bounded)
//   O += P·V'            via 4x V_WMMA_F32_16X16X4_F32 (K=16)
// where V' has V in columns 0..7, ONES in column 8, zeros in 9..15 -> the
// softmax denominator accumulates in O column 8 on the matrix pipe (no
// cross-lane reductions / ds_bpermute in the hot loop).
// ---------------------------------------------------------------------------
__global__ __launch_bounds__(32 * WAVES_PER_BLOCK) void attn_kernel(
    const float* __restrict__ q_ws, const float* __restrict__ k_ws,
    const float* __restrict__ vT_ws, const float* __restrict__ adj,
    const float* __restrict__ geo_w, float* __restrict__ attn_out)
{
    __shared__ float plds[WAVES_PER_BLOCK * 16 * 18];

    const int wave = threadIdx.x >> 5;
    const int lane = threadIdx.x & 31;
    const int g    = blockIdx.x * WAVES_PER_BLOCK + wave; // tile id in [0, B*H*128)
    const int bh   = g >> 7;           // / (N/16)
    const int tile = g & 127;
    const int rowbase = tile * 16;
    const int half = lane >> 4;        // 0 or 1
    const int c16  = lane & 15;        // 0..15
    const int hdc  = c16 & 7;          // clamped head-dim index for V loads
    const float selk = (c16 < 8)  ? 1.0f : 0.0f;  // keep real V columns
    const float sel1 = (c16 == 8) ? 1.0f : 0.0f;  // ones column -> row sums
    float* P = plds + wave * (16 * 18);

    const float gw    = geo_w[0];
    const float blend = 1.0f / (1.0f + __expf(-gw));
    const float a1c   = (1.0f - blend) * 0.35355339059f;  // (1-blend)/sqrt(HD)
    const float a2c   = blend * 5.0f;

    // Q fragments (A-matrix): lane m = c16, k = 2*half + j  (+4 for 2nd slice)
    const float* qbase = q_ws + ((size_t)bh * N_ + rowbase + c16) * HD_ + 2 * half;
    const v2f qa0 = *(const v2f*)(qbase);
    const v2f qa1 = *(const v2f*)(qbase + 4);

    v8f O = {0.f, 0.f, 0.f, 0.f, 0.f, 0.f, 0.f, 0.f};

    const float* krow = k_ws  + (size_t)bh * N_ * HD_;
    const float* vrow = vT_ws + (size_t)bh * HD_ * N_ + (size_t)hdc * N_;
    const float* arow = adj   + (size_t)(rowbase + 8 * half) * N_ + c16;

    for (int tb = 0; tb < 128; ++tb) {
        const int colbase = tb * 16;

        // K fragments (B-matrix): lane n = c16, k = 2*half + j
        const float* kb = krow + (size_t)(colbase + c16) * HD_ + 2 * half;
        const v2f kb0 = *(const v2f*)(kb);
        const v2f kb1 = *(const v2f*)(kb + 4);

        v8f S = {0.f, 0.f, 0.f, 0.f, 0.f, 0.f, 0.f, 0.f};
        S = __builtin_amdgcn_wmma_f32_16x16x4_f32(false, qa0, false, kb0,
                                                  (short)0, S, false, false);
        S = __builtin_amdgcn_wmma_f32_16x16x4_f32(false, qa1, false, kb1,
                                                  (short)0, S, false, false);

        // prefetch next adjacency tile into cache
        if (tb + 1 < 128)
            __builtin_prefetch(arow + colbase + 16, 0, 1);

        // P = exp(a1*S + a2*adj), staged to LDS in A-matrix order
        // (C/D layout: VGPR r -> row r + 8*half, lane c16 -> column)
#pragma unroll
        for (int r = 0; r < 8; ++r) {
            const float adjv = arow[(size_t)r * N_ + colbase];
            const float pv   = __expf(fmaf(a2c, adjv, a1c * S[r]));
            P[(r + 8 * half) * 18 + c16] = pv;
        }

        // O += P·V'  (4 K-slices of 4). Wave-private LDS: no barrier needed.
#pragma unroll
        for (int s = 0; s < 4; ++s) {
            const v2f pa = *(const v2f*)(P + c16 * 18 + 4 * s + 2 * half);
            v2f vb = *(const v2f*)(vrow + colbase + 4 * s + 2 * half);
            vb[0] = fmaf(vb[0], selk, sel1);   // col 8 -> 1.0, cols 9..15 -> 0
            vb[1] = fmaf(vb[1], selk, sel1);
            O = __builtin_amdgcn_wmma_f32_16x16x4_f32(false, pa, false, vb,
                                                      (short)0, O, false, false);
        }
    }

    // normalize by row sums carried in O column 8, scatter to [B, N, D]
    const int srclane = (lane & 16) + 8;       // lane holding this half's sums
    const int b = bh >> 2;
    const int h = bh & 3;
#pragma unroll
    for (int r = 0; r < 8; ++r) {
        const float l = __shfl(O[r], srclane, 32);
        const float o = O[r] / l;
        if (c16 < 8)
            attn_out[(size_t)(b * N_ + rowbase + r + 8 * half) * D_ +
                     h * HD_ + c16] = o;
    }
}

// ---------------------------------------------------------------------------
// Kernel 3: out = LN(attn @ Wout + bout + x) * gamma + beta
// One wave32 per row (D = 32 = wave width): shuffle broadcast + butterflies.
// ---------------------------------------------------------------------------
__global__ __launch_bounds__(256) void out_ln_kernel(
    const float* __restrict__ attn, const float* __restrict__ x,
    const float* __restrict__ Wout, const float* __restrict__ bout,
    const float* __restrict__ gamma, const float* __restrict__ beta,
    float* __restrict__ out)
{
    const int row = blockIdx.x * 8 + threadIdx.y;  // [0, B*N)
    const int t   = threadIdx.x;                   // 0..31

    const float a = attn[(size_t)row * D_ + t];
    float y = bout[t] + x[(size_t)row * D_ + t];
#pragma unroll
    for (int d = 0; d < 32; ++d)
        y = fmaf(__shfl(a, d, 32), Wout[d * D_ + t], y);

    float s = y;
    s += __shfl_xor(s, 16, 32); s += __shfl_xor(s, 8, 32);
    s += __shfl_xor(s, 4, 32);  s += __shfl_xor(s, 2, 32);
    s += __shfl_xor(s, 1, 32);
    const float mu = s * (1.0f / 32.0f);

    const float dv = y - mu;
    float vs = dv * dv;
    vs += __shfl_xor(vs, 16, 32); vs += __shfl_xor(vs, 8, 32);
    vs += __shfl_xor(vs, 4, 32);  vs += __shfl_xor(vs, 2, 32);
    vs += __shfl_xor(vs, 1, 32);
    const float var = vs * (1.0f / 32.0f);

    out[(size_t)row * D_ + t] = dv * rsqrtf(var + 1e-5f) * gamma[t] + beta[t];
}

// ---------------------------------------------------------------------------
extern "C" void kernel_launch(void* const* d_in, const int* in_sizes, int n_in,
                              void* d_out, int out_size, void* d_ws, size_t ws_size,
                              hipStream_t stream)
{
    const float* x     = (const float*)d_in[0];
    const float* adj   = (const float*)d_in[1];
    const float* geo   = (const float*)d_in[2];
    const float* Wqkv  = (const float*)d_in[3];
    const float* bqkv  = (const float*)d_in[4];
    const float* Wout  = (const float*)d_in[5];
    const float* bout  = (const float*)d_in[6];
    const float* gamma = (const float*)d_in[7];
    const float* beta  = (const float*)d_in[8];
    float* out = (float*)d_out;

    const size_t REG = (size_t)B_ * H_ * N_ * HD_;   // 524288 floats (2 MB)
    float* q_ws    = (float*)d_ws;
    float* k_ws    = q_ws + REG;
    float* vT_ws   = k_ws + REG;
    float* attn_ws = vT_ws + REG;

    dim3 blk(32, 8);
    qkv_kernel<<<(B_ * N_) / 8, blk, 0, stream>>>(x, Wqkv, bqkv,
                                                  q_ws, k_ws, vT_ws);

    const int tiles = B_ * H_ * (N_ / 16);           // 4096 waves
    attn_kernel<<<tiles / WAVES_PER_BLOCK, 32 * WAVES_PER_BLOCK, 0, stream>>>(
        q_ws, k_ws, vT_ws, adj, geo, attn_ws);

    out_ln_kernel<<<(B_ * N_) / 8, blk, 0, stream>>>(attn_ws, x, Wout, bout,
                                                     gamma, beta, out);
}